// Grapher_6356551598738
// MI455X (gfx1250) — compile-verified
//
#include <hip/hip_runtime.h>
#include <hip/hip_bf16.h>

// ---------------------------------------------------------------------------
// Grapher collapses algebraically to: out[n,c,p] = relu(x[n,c,p] + B2[c])
// where B2[c] = s4[c] * sum_o w4[c,o] * relu(cn[o]) + c4[c].
//   - inter_part_mr ignores input values (constant from row index),
//   - intra_part_mr uses zeros_like -> fc3 & first residual branch are dead,
//   - gconv of zeros + BN + ReLU == relu(cn), constant over batch and p,
//   - fc4 of a constant is a 64x128 matvec -> 64 scalars.
// So: tiny constant precompute (done with V_WMMA_F32_16X16X4_F32), then a
// 126 MB bandwidth-bound streaming pass (~5.4 us at 23.3 TB/s).
// ---------------------------------------------------------------------------

typedef __attribute__((ext_vector_type(2))) float v2f;
typedef __attribute__((ext_vector_type(4))) float v4f;
typedef __attribute__((ext_vector_type(8))) float v8f;

// ---------------------------------------------------------------------------
// Kernel A: B2[c] = s4[c] * (w4 . relu(cn))[c] + c4[c]   (c = 0..63)
// One workgroup, 4 waves. Wave w computes M-tile rows [16w, 16w+16) via
// 32 chained V_WMMA_F32_16X16X4_F32 (K = 128 in steps of 4).
//
// fp32 16x16x4 A layout (ISA 7.12.2): lanes 0-15 hold M=lane, reg0=K0,
// reg1=K1; lanes 16-31 hold M=lane-16, reg0=K2, reg1=K3. B mirrors with
// N=lane%16; we broadcast the vector across all N columns so every column
// of D equals the matvec result regardless of the exact N mapping.
// C/D layout: lane L, vgpr v -> M = v + (L>=16 ? 8 : 0), N = L%16.
// ---------------------------------------------------------------------------
__global__ __launch_bounds__(128) void grapher_bias_wmma(
    const float* __restrict__ cn,   // 128
    const float* __restrict__ w4,   // 64 x 128 row-major
    const float* __restrict__ s4,   // 64
    const float* __restrict__ c4,   // 64
    float* __restrict__ bias)       // out: 64 (in d_ws)
{
    const unsigned lane  = threadIdx.x & 31u;
    const unsigned wave  = threadIdx.x >> 5;        // 0..3 -> M tile
    const unsigned mrow  = wave * 16u + (lane & 15u);
    const unsigned khalf = (lane >> 4) << 1;        // 0 (lanes 0-15) / 2 (16-31)

    const float* __restrict__ wrow = w4 + mrow * 128u;

    v8f acc = {};
#pragma unroll
    for (unsigned kk = 0u; kk < 128u; kk += 4u) {
        const unsigned k = kk + khalf;
        v2f a, b;
        a.x = wrow[k];
        a.y = wrow[k + 1u];
        float u0 = cn[k];
        float u1 = cn[k + 1u];
        b.x = u0 > 0.0f ? u0 : 0.0f;   // relu(cn)
        b.y = u1 > 0.0f ? u1 : 0.0f;
        // (neg_a, A, neg_b, B, c_mod, C, reuse_a, reuse_b)
        acc = __builtin_amdgcn_wmma_f32_16x16x4_f32(
            false, a, false, b, (short)0, acc, false, false);
    }

    // Every N column of D holds the matvec; let lanes with N==0 write out.
    if ((lane & 15u) == 0u) {
        const unsigned mbase = wave * 16u + ((lane >> 4) << 3); // +0 or +8
#pragma unroll
        for (int v = 0; v < 8; ++v) {
            const unsigned c = mbase + (unsigned)v;
            bias[c] = acc[v] * s4[c] + c4[c];
        }
    }
}

// ---------------------------------------------------------------------------
// Kernel B: out[i] = relu(x[i] + B2[(i/15) & 63])  (flat (BT,64,15,1) layout)
// float4 per thread, bias table in LDS, non-temporal stores (write-once).
// ---------------------------------------------------------------------------
__global__ __launch_bounds__(256) void grapher_bias_relu(
    const float* __restrict__ x,
    const float* __restrict__ bias,   // 64 floats (d_ws)
    float* __restrict__ out,
    unsigned n4,                      // count of float4 elements
    unsigned n)                       // total scalar elements
{
    __shared__ float b2[64];
    if (threadIdx.x < 64u) b2[threadIdx.x] = bias[threadIdx.x];
    __syncthreads();

    const unsigned t = blockIdx.x * blockDim.x + threadIdx.x;
    if (t < n4) {
        const unsigned j = t * 4u;
        v4f v = ((const v4f*)x)[t];
        v4f r;
        r.x = v.x + b2[((j      ) / 15u) & 63u];
        r.y = v.y + b2[((j + 1u) / 15u) & 63u];
        r.z = v.z + b2[((j + 2u) / 15u) & 63u];
        r.w = v.w + b2[((j + 3u) / 15u) & 63u];
        r.x = r.x > 0.0f ? r.x : 0.0f;
        r.y = r.y > 0.0f ? r.y : 0.0f;
        r.z = r.z > 0.0f ? r.z : 0.0f;
        r.w = r.w > 0.0f ? r.w : 0.0f;
        __builtin_nontemporal_store(r, ((v4f*)out) + t);
    }
    // Scalar tail (empty for this problem: 15,728,640 % 4 == 0).
    if (t == 0u) {
        for (unsigned i = n4 * 4u; i < n; ++i) {
            float r = x[i] + b2[(i / 15u) & 63u];
            out[i] = r > 0.0f ? r : 0.0f;
        }
    }
}

// ---------------------------------------------------------------------------
// Input order (setup_inputs): 0:x 1:w1 2:s1 3:c1 4:wi 5:si 6:ci 7:w2 8:s2
// 9:c2 10:w3 11:s3 12:c3 13:wn 14:sn 15:cn 16:w4 17:s4 18:c4
// ---------------------------------------------------------------------------
extern "C" void kernel_launch(void* const* d_in, const int* in_sizes, int n_in,
                              void* d_out, int out_size, void* d_ws, size_t ws_size,
                              hipStream_t stream) {
    const float* x  = (const float*)d_in[0];
    const float* cn = (const float*)d_in[15];
    const float* w4 = (const float*)d_in[16];
    const float* s4 = (const float*)d_in[17];
    const float* c4 = (const float*)d_in[18];
    float* out  = (float*)d_out;
    float* bias = (float*)d_ws;   // 64 floats of scratch

    // Tiny constant fold: 64x128 matvec via fp32 WMMA (one workgroup).
    grapher_bias_wmma<<<dim3(1), dim3(128), 0, stream>>>(cn, w4, s4, c4, bias);

    // Streaming bias + ReLU over 15,728,640 floats (126 MB total traffic).
    const unsigned n  = (unsigned)in_sizes[0];
    const unsigned n4 = n / 4u;
    const unsigned threads = 256u;
    const unsigned blocks  = (n4 + threads - 1u) / threads;
    grapher_bias_relu<<<dim3(blocks), dim3(threads), 0, stream>>>(
        x, bias, out, n4, n);
}